// AttentionEncoder_36558761623585
// MI455X (gfx1250) — compile-verified
//
#include <hip/hip_runtime.h>
#include <cmath>
#include <cstdint>

// ---------------------------------------------------------------------------
// GRU encoder for MI455X (gfx1250, wave32, WMMA + async global->LDS staging).
// fp32 matrix path: V_WMMA_F32_16X16X4_F32; A(16x4) and B(4x16) K-major so
// each lane reads one contiguous float2 per operand (ds_load_b64 from LDS).
// ---------------------------------------------------------------------------

typedef __attribute__((ext_vector_type(2))) float v2f;
typedef __attribute__((ext_vector_type(8))) float v8f;
typedef int i4_t __attribute__((vector_size(16)));   // matches builtin param type
typedef __attribute__((address_space(1))) i4_t* g_i4p;   // global int4*
typedef __attribute__((address_space(3))) i4_t* l_i4p;   // LDS int4*

#define S_LEN   256
#define BATCH_N 64
#define EMBED_N 1024
#define HID     1024
#define GATES   3072
#define PAD_TOK 0
#define KB      32          // K-panel depth
#define LDS_STR 36          // panel stride (floats): 36 = 4*9 mod 64 -> no bank conflicts

#if __has_builtin(__builtin_amdgcn_global_load_async_to_lds_b128)
#define ASYNC_LDS 1
#endif

static __device__ __forceinline__ void cp16_g2l(float* lds_dst, const float* gsrc) {
#ifdef ASYNC_LDS
  __builtin_amdgcn_global_load_async_to_lds_b128(
      (g_i4p)(uintptr_t)gsrc,
      (l_i4p)(unsigned)(uintptr_t)lds_dst,
      0, 0);
#else
  *(float4*)lds_dst = *(const float4*)gsrc;
#endif
}

static __device__ __forceinline__ void async_panel_wait() {
#ifdef ASYNC_LDS
#if __has_builtin(__builtin_amdgcn_s_wait_asynccnt)
  __builtin_amdgcn_s_wait_asynccnt(0);
#else
  asm volatile("s_wait_asynccnt 0" ::: "memory");
#endif
#endif
}

static __device__ __forceinline__ v8f wmma_f32(v2f a, v2f b, v8f c) {
  return __builtin_amdgcn_wmma_f32_16x16x4_f32(false, a, false, b,
                                               (short)0, c, false, false);
}

// Grid-wide barrier: monotonically increasing counter, release/acquire scope.
static __device__ __forceinline__ void grid_barrier(unsigned* cnt, unsigned target) {
  __syncthreads();
  if (threadIdx.x == 0) {
    __hip_atomic_fetch_add(cnt, 1u, __ATOMIC_RELEASE, __HIP_MEMORY_SCOPE_AGENT);
    while (__hip_atomic_load(cnt, __ATOMIC_ACQUIRE, __HIP_MEMORY_SCOPE_AGENT) < target)
      __builtin_amdgcn_s_sleep(1);
  }
  __syncthreads();
}

// ---------------------------------------------------------------------------
__global__ void valid_lens_kernel(const int* __restrict__ src, float* __restrict__ vl) {
  const int b = threadIdx.x;
  if (b >= BATCH_N) return;
  int idx = -1;
  for (int s = 0; s < S_LEN; ++s)
    if (src[b * S_LEN + s] == PAD_TOK) idx = s;
  vl[b] = (idx >= 0) ? (float)idx : (float)S_LEN;
}

// ---------------------------------------------------------------------------
__global__ __launch_bounds__(256)
void embed_gather(const int* __restrict__ src, const float* __restrict__ emb,
                  float* __restrict__ x) {
  const int sb = blockIdx.x;       // s*B + b
  const int s  = sb >> 6;
  const int b  = sb & 63;
  const int tok = src[b * S_LEN + s];
  const float4* e  = (const float4*)(emb + (size_t)tok * EMBED_N);
  float4*       xo = (float4*)(x + (size_t)sb * EMBED_N);
  xo[threadIdx.x] = e[threadIdx.x];        // 256 thr * 16B = 4KB row
}

// ---------------------------------------------------------------------------
// Y[m,n] = sum_k X[m,k]*W[n,k] + bias[n].  X: MxK row-major, W: NxK row-major.
// Block tile 64Mx128N, 8 waves (2x4), wave tile 32x32 (2x2 WMMA tiles).
// K-panels of 32 staged in LDS (double-buffered, async global->LDS).
// ---------------------------------------------------------------------------
__global__ __launch_bounds__(256)
void gemm_rowrow_bias(const float* __restrict__ X, const float* __restrict__ W,
                      const float* __restrict__ bias, float* __restrict__ Y,
                      int M, int N, int K) {
  __shared__ float sA[2][64 * LDS_STR];    //  18 KB
  __shared__ float sB[2][128 * LDS_STR];   //  36 KB

  const int t    = threadIdx.x;
  const int warp = t >> 5;
  const int lane = t & 31;
  const int lr   = lane & 15;
  const int kh   = lane >> 4;              // K half-select
  const int wm   = warp >> 2;              // 0..1
  const int wn   = warp & 3;               // 0..3
  const int mblk = blockIdx.x * 64;
  const int nblk = blockIdx.y * 128;

  // stage one K-panel (A: 64x32 = 512 16B-chunks, B: 128x32 = 1024 chunks)
  auto stage = [&](int buf, int kb) {
#pragma unroll
    for (int i = 0; i < 2; ++i) {
      const int ch = t + i * 256, r = ch >> 3, c = ch & 7;
      cp16_g2l(&sA[buf][r * LDS_STR + c * 4],
               X + (size_t)(mblk + r) * K + kb + c * 4);
    }
#pragma unroll
    for (int i = 0; i < 4; ++i) {
      const int ch = t + i * 256, r = ch >> 3, c = ch & 7;
      cp16_g2l(&sB[buf][r * LDS_STR + c * 4],
               W + (size_t)(nblk + r) * K + kb + c * 4);
    }
  };

  v8f c00 = {}, c01 = {}, c10 = {}, c11 = {};
  stage(0, 0);
  int buf = 0;
  for (int kb = 0; kb < K; kb += KB) {
    async_panel_wait();
    __syncthreads();                       // panel `buf` ready; prev reads done
    if (kb + KB < K) stage(buf ^ 1, kb + KB);

    const float* la0 = &sA[buf][(wm * 32 + lr) * LDS_STR];
    const float* la1 = la0 + 16 * LDS_STR;
    const float* lb0 = &sB[buf][(wn * 32 + lr) * LDS_STR];
    const float* lb1 = lb0 + 16 * LDS_STR;
#pragma unroll
    for (int k = 0; k < KB; k += 4) {
      const int kk = k + 2 * kh;
      v2f a0 = *(const v2f*)(la0 + kk);
      v2f a1 = *(const v2f*)(la1 + kk);
      v2f b0 = *(const v2f*)(lb0 + kk);
      v2f b1 = *(const v2f*)(lb1 + kk);
      c00 = wmma_f32(a0, b0, c00);
      c01 = wmma_f32(a0, b1, c01);
      c10 = wmma_f32(a1, b0, c10);
      c11 = wmma_f32(a1, b1, c11);
    }
    buf ^= 1;
  }

  // D layout: VGPR r -> row tile_m0 + r + 8*(lane>=16), col tile_n0 + (lane&15)
  const int col0 = nblk + wn * 32 + lr, col1 = col0 + 16;
  const float bb0 = bias[col0], bb1 = bias[col1];
  const int r0 = mblk + wm * 32 + 8 * kh, r1 = r0 + 16;
#pragma unroll
  for (int r = 0; r < 8; ++r) {
    Y[(size_t)(r0 + r) * N + col0] = c00[r] + bb0;
    Y[(size_t)(r0 + r) * N + col1] = c01[r] + bb1;
    Y[(size_t)(r1 + r) * N + col0] = c10[r] + bb0;
    Y[(size_t)(r1 + r) * N + col1] = c11[r] + bb1;
  }
}

// ---------------------------------------------------------------------------
// Persistent GRU scan. 48 blocks x 8 waves = 384 waves = 768 tiles / 2.
// All 8 waves of a block share one 16-row h panel (same batch tile) -> stage
// it in LDS per K-panel (double-buffered async). Wh streams from L2.
// ---------------------------------------------------------------------------
__global__ __launch_bounds__(256)
void gru_scan(const float* __restrict__ xp, const float* __restrict__ Wh,
              const float* __restrict__ bh, float* __restrict__ h,
              float* __restrict__ gh, float* __restrict__ out,
              float* __restrict__ stateOut, unsigned* __restrict__ sync,
              int nblocks) {
  __shared__ float sH[2][16 * LDS_STR];    // 4.6 KB

  const int t    = threadIdx.x;
  const int warp = t >> 5;
  const int lane = t & 31;
  const int lr   = lane & 15;
  const int kh   = lane >> 4;
  const int g    = blockIdx.x * 8 + warp;  // 0..383
  const int m0   = (g / 96) * 16;          // uniform across the block (96%8==0)
  const int n0   = (g % 96) * 32;
  const int gtid = blockIdx.x * 256 + t;
  const int gsz  = nblocks * 256;

  const float* B0 = Wh + (size_t)(n0 + lr) * HID;
  const float* B1 = B0 + (size_t)16 * HID;
  const int col0 = n0 + lr, col1 = col0 + 16;
  const float bb0 = bh[col0], bb1 = bh[col1];

  // stage 16x32 h panel: 128 16B-chunks, threads 0..127 (waves 0-3, uniform)
  auto stageH = [&](int buf, int kb) {
    if (t < 128) {
      const int r = t >> 3, c = t & 7;
      cp16_g2l(&sH[buf][r * LDS_STR + c * 4],
               h + (size_t)(m0 + r) * HID + kb + c * 4);
    }
  };

  unsigned bar = 0;
  for (int s = 0; s < S_LEN; ++s) {
    // ---- phase A: gh tile = h x Wh^T + bh -------------------------------
    v8f c0 = {}, c1 = {};
    stageH(0, 0);
    int buf = 0;
    for (int kb = 0; kb < HID; kb += KB) {
      async_panel_wait();
      __syncthreads();
      if (kb + KB < HID) stageH(buf ^ 1, kb + KB);
      const float* la = &sH[buf][lr * LDS_STR];
#pragma unroll
      for (int k = 0; k < KB; k += 4) {
        const int kk = k + 2 * kh;
        v2f a  = *(const v2f*)(la + kk);
        v2f b0 = *(const v2f*)(B0 + kb + kk);
        v2f b1 = *(const v2f*)(B1 + kb + kk);
        c0 = wmma_f32(a, b0, c0);
        c1 = wmma_f32(a, b1, c1);
      }
      buf ^= 1;
    }
    const int r0 = m0 + 8 * kh;
#pragma unroll
    for (int r = 0; r < 8; ++r) {
      gh[(size_t)(r0 + r) * GATES + col0] = c0[r] + bb0;
      gh[(size_t)(r0 + r) * GATES + col1] = c1[r] + bb1;
    }
    bar++; grid_barrier(sync, (unsigned)nblocks * bar);

    // ---- phase B: gates -------------------------------------------------
    const float* xps  = xp + (size_t)s * (BATCH_N * GATES);
    float*       outs = out + (size_t)s * (BATCH_N * HID);
    for (int i = gtid; i < BATCH_N * HID; i += gsz) {
      const int b = i >> 10;
      const int j = i & 1023;
      const float* xrow = xps + (size_t)b * GATES;
      const float* grow = gh  + (size_t)b * GATES;
      const float r = 1.0f / (1.0f + __expf(-(xrow[j]       + grow[j])));
      const float z = 1.0f / (1.0f + __expf(-(xrow[HID + j] + grow[HID + j])));
      const float n = tanhf(xrow[2 * HID + j] + r * grow[2 * HID + j]);
      const float hnew = (1.0f - z) * n + z * h[i];
      h[i]    = hnew;
      outs[i] = hnew;
    }
    bar++; grid_barrier(sync, (unsigned)nblocks * bar);
  }

  for (int i = gtid; i < BATCH_N * HID; i += gsz) stateOut[i] = h[i];
}

// ---------------------------------------------------------------------------
__global__ __launch_bounds__(256)
void init_layer(float* __restrict__ h, unsigned* __restrict__ sync) {
  const int i = blockIdx.x * blockDim.x + threadIdx.x;
  if (i < BATCH_N * HID) h[i] = 0.0f;
  if (i == 0)
    __hip_atomic_store(sync, 0u, __ATOMIC_RELEASE, __HIP_MEMORY_SCOPE_AGENT);
}

// ---------------------------------------------------------------------------
extern "C" void kernel_launch(void* const* d_in, const int* in_sizes, int n_in,
                              void* d_out, int out_size, void* d_ws, size_t ws_size,
                              hipStream_t stream) {
  const int*   src = (const int*)  d_in[0];   // (B,S)
  const float* emb = (const float*)d_in[1];   // (V,E)
  const float* Wx  = (const float*)d_in[2];   // (2,3H,H)
  const float* Wh  = (const float*)d_in[3];   // (2,3H,H)
  const float* bx  = (const float*)d_in[4];   // (2,3H)
  const float* bh  = (const float*)d_in[5];   // (2,3H)
  float* out = (float*)d_out;

  // workspace layout (floats)
  float*    x    = (float*)d_ws;            // S*B*E      = 16,777,216
  float*    xp   = x + 16777216;            // S*B*3H     = 50,331,648
  float*    h    = x + 67108864;            // B*H        =     65,536
  float*    gh   = x + 67174400;            // B*3H       =    196,608
  unsigned* sync = (unsigned*)(x + 67371008);

  float* x_out = out;                        // (S,B,H)
  float* state = out + 16777216;             // (2,B,H)
  float* vlens = out + 16777216 + 131072;    // (B,)

  valid_lens_kernel<<<1, 64, 0, stream>>>(src, vlens);
  embed_gather<<<S_LEN * BATCH_N, 256, 0, stream>>>(src, emb, x);

  const dim3 ggrid(16384 / 64, GATES / 128);
  const size_t wstep = (size_t)GATES * HID;

  // layer 0: out overwrites x in place (xp already holds x*Wx^T)
  init_layer<<<256, 256, 0, stream>>>(h, sync);
  gemm_rowrow_bias<<<ggrid, 256, 0, stream>>>(x, Wx, bx, xp, 16384, GATES, HID);
  gru_scan<<<48, 256, 0, stream>>>(xp, Wh, bh, h, gh, x, state, sync, 48);

  // layer 1
  init_layer<<<256, 256, 0, stream>>>(h, sync);
  gemm_rowrow_bias<<<ggrid, 256, 0, stream>>>(x, Wx + wstep, bx + GATES, xp,
                                              16384, GATES, HID);
  gru_scan<<<48, 256, 0, stream>>>(xp, Wh + wstep, bh + GATES, h, gh, x_out,
                                   state + BATCH_N * HID, sync, 48);
}